// MeanSquaredError2___15221364097462
// MI455X (gfx1250) — compile-verified
//
#include <hip/hip_runtime.h>
#include <hip/hip_bf16.h>
#include <math.h>

#define NJ   14
#define COLS 14
#define NCH  18   // NJ + 4 group channels
#define RAD  4

typedef __attribute__((ext_vector_type(2))) float v2f;
typedef __attribute__((ext_vector_type(8))) float v8f;

// Wave32 XOR-shuffle: ds_swizzle group-of-32 mode, offset = {xor[14:10], or[9:5]=0, and[4:0]=0x1f}.
// One DS op, no index-VGPR setup (vs __shfl_xor's mbcnt/xor/cmp/cndmask/bpermute sequence).
#define SWZ_XOR(x, m) \
  __int_as_float(__builtin_amdgcn_ds_swizzle(__float_as_int(x), (((m) << 10) | 0x1f)))

// Normalized 1-D Gaussian taps, sigma=1, radius=4 (compile-time constants; taps are
// symmetric: k[i] == k[8-i]).
#define KT0 1.3383063e-4f
#define KT1 4.4318620e-3f
#define KT2 5.3991131e-2f
#define KT3 2.4197146e-1f
#define KT4 3.9894351e-1f

// Reflect index for jnp.pad(..., mode='symmetric'): -1 -> 0, -2 -> 1, 14 -> 13, 15 -> 12, ...
__device__ __forceinline__ int reflect_idx(int i) {
  if (i < 0) return -i - 1;
  if (i > COLS - 1) return 2 * COLS - 1 - i;
  return i;
}

// W[a][p] = sum_{d=-4..4} k1d[d+4] * [reflect(a+d) == p]
// (1-D Gaussian blur operator with symmetric padding, as a 14x14 matrix)
__device__ __forceinline__ float wentry(int a, int p) {
  const float k1d[2 * RAD + 1] = {KT0, KT1, KT2, KT3, KT4, KT3, KT2, KT1, KT0};
  float w = 0.0f;
#pragma unroll
  for (int d = -RAD; d <= RAD; ++d) {
    int i = reflect_idx(a + d);
    w += (i == p) ? k1d[d + RAD] : 0.0f;
  }
  return w;
}

__global__ __launch_bounds__(256) void heatmap_mse_kernel(
    const float* __restrict__ h, const float* __restrict__ t,
    const float* __restrict__ v, float* __restrict__ partials, int nch_total) {
  const int wave = threadIdx.x >> 5;
  const int lane = threadIdx.x & 31;
  const int ch   = blockIdx.x * 8 + wave;

  __shared__ float s_sum[8];
  __shared__ float s_cnt[8];

  float wave_sum = 0.0f;
  float wave_cnt = 0.0f;

  if (ch < nch_total) {
    const int b = ch / NCH;
    const int c = ch % NCH;

    // --- point extraction (wave-uniform scalar work, done redundantly per lane) ---
    int px0 = 0, py0 = 0, px1 = 0, py1 = 0, px2 = 0, py2 = 0;
    int npts = 0;
    bool active = false;
    const int tb = b * (NJ * 2);

    if (c < NJ) {
      const int j = c;
      const bool vis = v[tb + j * 2] > 0.5f;
      active = vis;
      if (vis) {
        int xi = (int)(t[tb + j * 2 + 0] * (float)COLS);
        int yi = (int)(t[tb + j * 2 + 1] * (float)COLS);
        px0 = min(max(xi, 0), COLS - 1);
        py0 = min(max(yi, 0), COLS - 1);
        npts = 1;
      }
    } else {
      const int grp = c - NJ;
#pragma unroll
      for (int jj = 0; jj < 3; ++jj) {
        const int j = grp * 3 + jj;
        const bool vis = v[tb + j * 2] > 0.5f;
        if (vis) {
          active = true;
          int xi = (int)(t[tb + j * 2 + 0] * (float)COLS);
          int yi = (int)(t[tb + j * 2 + 1] * (float)COLS);
          xi = min(max(xi, 0), COLS - 1);
          yi = min(max(yi, 0), COLS - 1);
          // scatter-max collapses duplicates: dedupe
          const bool dup = (npts > 0 && px0 == xi && py0 == yi) ||
                           (npts > 1 && px1 == xi && py1 == yi);
          if (!dup) {
            if (npts == 0)      { px0 = xi; py0 = yi; }
            else if (npts == 1) { px1 = xi; py1 = yi; }
            else                { px2 = xi; py2 = yi; }
            ++npts;
          }
        }
      }
    }

    if (active) {  // wave-uniform branch: inactive channels skip their h traffic entirely
      const int  half   = lane >> 4;   // 0: K slots 0,1 / rows 0..7 ; 1: K slots 2,3 / rows 8..15
      const int  rc     = lane & 15;   // A row (y) and B column (x)
      const bool col_ok = (rc < COLS);

      // Branchless A/B fragment build:
      //   A[y][k] = W[y][py_k] ; B[k][x] = W[x][px_k] ; zero outside 14x14 and k >= npts.
      //   half==0 holds K slots {0,1}; half==1 holds K slots {2,3}; slot 3 always empty.
      const int  pA0  = half ? py2 : py0;
      const int  pB0  = half ? px2 : px0;
      const bool s0ok = col_ok && (half ? (npts > 2) : (npts > 0));
      const bool s1ok = col_ok && (half == 0) && (npts > 1);

      const float a0 = s0ok ? wentry(rc, pA0) : 0.0f;
      const float b0 = s0ok ? wentry(rc, pB0) : 0.0f;
      const float a1 = s1ok ? wentry(rc, py1) : 0.0f;
      const float b1 = s1ok ? wentry(rc, px1) : 0.0f;

      v2f afrag; afrag.x = a0; afrag.y = a1;
      v2f bfrag; bfrag.x = b0; bfrag.y = b1;

      // g(16x16) = A(16x4) * B(4x16): entire blurred map in one matrix op.
      v8f cfrag = {};
      cfrag = __builtin_amdgcn_wmma_f32_16x16x4_f32(
          /*neg_a=*/false, afrag, /*neg_b=*/false, bfrag,
          /*c_mod=*/(short)0, cfrag, /*reuse_a=*/false, /*reuse_b=*/false);

      // Branchless, always-in-bounds (clamped) h loads; streamed once -> non-temporal
      // (h is 231 MB > 192 MB L2; NT keeps L2 clean for t/v/partials).
      const float* __restrict__ hch = h + (size_t)ch * (COLS * COLS);
      const int xc = min(rc, COLS - 1);
      float hv[8];
      float lmin = 3.0e38f, lmax = -3.0e38f;
#pragma unroll
      for (int r = 0; r < 8; ++r) {
        const int  y     = r + half * 8;
        const int  yc    = min(y, COLS - 1);
        const bool valid = col_ok && (y < COLS);
        const float hraw = __builtin_nontemporal_load(hch + yc * COLS + xc);
        hv[r] = valid ? hraw : 0.0f;
        const float gval = cfrag[r];
        lmin = fminf(lmin, valid ? gval :  3.0e38f);
        lmax = fmaxf(lmax, valid ? gval : -3.0e38f);
      }
      // Wave32 min/max reduction via immediate-pattern ds_swizzle XOR shuffles.
      lmin = fminf(lmin, SWZ_XOR(lmin, 16)); lmax = fmaxf(lmax, SWZ_XOR(lmax, 16));
      lmin = fminf(lmin, SWZ_XOR(lmin,  8)); lmax = fmaxf(lmax, SWZ_XOR(lmax,  8));
      lmin = fminf(lmin, SWZ_XOR(lmin,  4)); lmax = fmaxf(lmax, SWZ_XOR(lmax,  4));
      lmin = fminf(lmin, SWZ_XOR(lmin,  2)); lmax = fmaxf(lmax, SWZ_XOR(lmax,  2));
      lmin = fminf(lmin, SWZ_XOR(lmin,  1)); lmax = fmaxf(lmax, SWZ_XOR(lmax,  1));

      const float denom = lmax - lmin;
      const float rden  = 1.0f / ((denom > 0.0f) ? denom : 1.0f);

      float acc = 0.0f;
#pragma unroll
      for (int r = 0; r < 8; ++r) {
        const int  y     = r + half * 8;
        const bool valid = col_ok && (y < COLS);
        const float mm   = (cfrag[r] - lmin) * rden;
        const float d    = valid ? (hv[r] - mm) : 0.0f;
        acc += d * d;
      }
      acc += SWZ_XOR(acc, 16);
      acc += SWZ_XOR(acc,  8);
      acc += SWZ_XOR(acc,  4);
      acc += SWZ_XOR(acc,  2);
      acc += SWZ_XOR(acc,  1);

      wave_sum = acc;
      wave_cnt = 1.0f;
    }
  }

  if (lane == 0) { s_sum[wave] = wave_sum; s_cnt[wave] = wave_cnt; }
  __syncthreads();
  if (threadIdx.x == 0) {
    float s = 0.0f, cc = 0.0f;
#pragma unroll
    for (int i = 0; i < 8; ++i) { s += s_sum[i]; cc += s_cnt[i]; }
    partials[2 * (size_t)blockIdx.x + 0] = s;
    partials[2 * (size_t)blockIdx.x + 1] = cc;
  }
}

// Deterministic final reduction: fixed iteration order + fixed LDS tree.
__global__ __launch_bounds__(256) void finalize_kernel(
    const float* __restrict__ partials, int n, float* __restrict__ out) {
  __shared__ float ssum[256];
  __shared__ float scnt[256];
  float s = 0.0f, c = 0.0f;
  for (int i = threadIdx.x; i < n; i += 256) {
    s += partials[2 * (size_t)i + 0];
    c += partials[2 * (size_t)i + 1];
  }
  ssum[threadIdx.x] = s;
  scnt[threadIdx.x] = c;
  __syncthreads();
  for (int step = 128; step > 0; step >>= 1) {
    if (threadIdx.x < step) {
      ssum[threadIdx.x] += ssum[threadIdx.x + step];
      scnt[threadIdx.x] += scnt[threadIdx.x + step];
    }
    __syncthreads();
  }
  if (threadIdx.x == 0) {
    out[0] = ssum[0] / (scnt[0] * (float)(COLS * COLS));
  }
}

extern "C" void kernel_launch(void* const* d_in, const int* in_sizes, int n_in,
                              void* d_out, int out_size, void* d_ws, size_t ws_size,
                              hipStream_t stream) {
  (void)n_in; (void)out_size; (void)ws_size;
  // setup_inputs order: os_ (unused by reference), h, t, v
  const float* h = (const float*)d_in[1];
  const float* t = (const float*)d_in[2];
  const float* v = (const float*)d_in[3];

  const int B       = in_sizes[1] / (NCH * COLS * COLS);
  const int nch     = B * NCH;
  const int nblocks = (nch + 7) / 8;   // 8 waves (channels) per 256-thread block
  float* partials   = (float*)d_ws;    // nblocks * 2 floats

  heatmap_mse_kernel<<<nblocks, 256, 0, stream>>>(h, t, v, partials, nch);
  finalize_kernel<<<1, 256, 0, stream>>>(partials, nblocks, (float*)d_out);
}